// EMD_Wrapper_27101243637906
// MI455X (gfx1250) — compile-verified
//
#include <hip/hip_runtime.h>
#include <math.h>

// ---------------------------------------------------------------------------
// Entropic-OT (Sinkhorn) EMD for B=16 batches of N=2048 3-D points.
// Cost tiles are recomputed in-register via V_WMMA_F32_16X16X4_F32
// (A = [p0,p1,p2,1], B = [-2q0,-2q1,-2q2,|q|^2 - dual]), so the 256 MB cost
// matrix never touches HBM/L2. Streaming (online) logsumexp per 16-row tile.
// ---------------------------------------------------------------------------

#define BATCH 16
#define N 2048
#define EPS 0.01f
#define INV_EPS 100.0f
#define N_ITERS 50
#define LOG_MARG -7.6246189861593985f   // -log(2048)

typedef __attribute__((ext_vector_type(2))) float v2f;
typedef __attribute__((ext_vector_type(8))) float v8f;

// Zero the dual potentials and the output accumulator (re-run every call:
// the harness replays the captured graph without re-poisoning buffers).
__global__ void emd_prep(float* __restrict__ f, float* __restrict__ g,
                         float* __restrict__ out) {
    int i = blockIdx.x * blockDim.x + threadIdx.x;
    if (i < BATCH * N) { f[i] = 0.0f; g[i] = 0.0f; }
    if (i < BATCH) out[i] = 0.0f;
}

// One Sinkhorn half-step:
//   dual_row_out[r] = EPS * (log_marg - LSE_c( (dual_col[c] - C_rc) / EPS ))
// where C_rc = |Prow_r|^2 + |Pcol_c|^2 - 2 Prow_r . Pcol_c.
// f-update: (Prow,Pcol,dual_col,out) = (x, y, g, f)
// g-update: (Prow,Pcol,dual_col,out) = (y, x, f, g)
// Block = 256 threads = 8 wave32; each wave owns one 16-row tile and streams
// all 128 16-col tiles through WMMA with an online logsumexp.
__global__ __launch_bounds__(256) void sinkhorn_half(
    const float* __restrict__ Prow, const float* __restrict__ Pcol,
    const float* __restrict__ dual_col, float* __restrict__ dual_row_out) {
    __shared__ float lcol[N * 4];   // per col: {-2q0, -2q1, -2q2, |q|^2 - dual}

    const int b   = blockIdx.y;
    const int tid = threadIdx.x;
    const float* pc = Pcol + (size_t)b * N * 3;
    const float* dc = dual_col + (size_t)b * N;
    for (int c = tid; c < N; c += 256) {
        float q0 = pc[c * 3 + 0], q1 = pc[c * 3 + 1], q2 = pc[c * 3 + 2];
        lcol[c * 4 + 0] = -2.0f * q0;
        lcol[c * 4 + 1] = -2.0f * q1;
        lcol[c * 4 + 2] = -2.0f * q2;
        lcol[c * 4 + 3] = (q0 * q0 + q1 * q1 + q2 * q2) - dc[c];
    }
    __syncthreads();

    const int lane  = tid & 31;
    const int half  = lane >> 4;          // 0: K=0..1 / rows M=k ; 1: K=2..3 / rows M=k+8
    const int l16   = lane & 15;
    const int itile = blockIdx.x * 8 + (tid >> 5);   // 0..127
    const int rowBase = itile * 16;

    const float* pr = Prow + (size_t)b * N * 3;

    // A operand: 16x4 f32, lane<16 -> (K0,K1), lane>=16 -> (K2,K3) of row M=l16
    v2f a;
    {
        int r = rowBase + l16;
        float p0 = pr[r * 3 + 0], p1 = pr[r * 3 + 1], p2 = pr[r * 3 + 2];
        a.x = half ? p2 : p0;
        a.y = half ? 1.0f : p1;
    }
    // Per-lane row constants |p|^2 for D rows M = k + 8*half (k = VGPR index)
    float rr[8];
#pragma unroll
    for (int k = 0; k < 8; ++k) {
        int r = rowBase + k + 8 * half;
        float p0 = pr[r * 3 + 0], p1 = pr[r * 3 + 1], p2 = pr[r * 3 + 2];
        rr[k] = p0 * p0 + p1 * p1 + p2 * p2;
    }

    float m[8], l[8];
#pragma unroll
    for (int k = 0; k < 8; ++k) { m[k] = -INFINITY; l[k] = 0.0f; }

    for (int jt = 0; jt < 128; ++jt) {
        int c = jt * 16 + l16;
        v2f bb;
        bb.x = lcol[c * 4 + 2 * half + 0];
        bb.y = lcol[c * 4 + 2 * half + 1];
        v8f d = {};
        // S = -2 p.q + |q|^2 - dual_c  =>  (dual_c - C)/EPS = -(S + |p|^2)/EPS
        d = __builtin_amdgcn_wmma_f32_16x16x4_f32(false, a, false, bb,
                                                  (short)0, d, false, false);
#pragma unroll
        for (int k = 0; k < 8; ++k) {
            float v  = -(d[k] + rr[k]) * INV_EPS;
            float mn = fmaxf(m[k], v);
            l[k] = l[k] * __expf(m[k] - mn) + __expf(v - mn);
            m[k] = mn;
        }
    }

    // Merge per-lane (m,l) across the 16 lanes holding one row (wave32 halves)
#pragma unroll
    for (int k = 0; k < 8; ++k) {
#pragma unroll
        for (int off = 1; off < 16; off <<= 1) {
            float om = __shfl_xor(m[k], off, 32);
            float ol = __shfl_xor(l[k], off, 32);
            float mn = fmaxf(m[k], om);
            l[k] = l[k] * __expf(m[k] - mn) + ol * __expf(om - mn);
            m[k] = mn;
        }
    }

    if (l16 == 0) {
        float* outp = dual_row_out + (size_t)b * N;
#pragma unroll
        for (int k = 0; k < 8; ++k) {
            float lse = m[k] + __logf(l[k]);
            outp[rowBase + k + 8 * half] = EPS * (LOG_MARG - lse);
        }
    }
}

// Final contraction: out[b] = sum_ij exp((f_i + g_j - C_ij)/EPS) * C_ij
__global__ __launch_bounds__(256) void emd_final(
    const float* __restrict__ X, const float* __restrict__ Y,
    const float* __restrict__ F, const float* __restrict__ G,
    float* __restrict__ out) {
    __shared__ float lcol[N * 4];
    __shared__ float blockAcc;

    const int b   = blockIdx.y;
    const int tid = threadIdx.x;
    if (tid == 0) blockAcc = 0.0f;

    const float* yb = Y + (size_t)b * N * 3;
    const float* gb = G + (size_t)b * N;
    for (int c = tid; c < N; c += 256) {
        float q0 = yb[c * 3 + 0], q1 = yb[c * 3 + 1], q2 = yb[c * 3 + 2];
        lcol[c * 4 + 0] = -2.0f * q0;
        lcol[c * 4 + 1] = -2.0f * q1;
        lcol[c * 4 + 2] = -2.0f * q2;
        lcol[c * 4 + 3] = (q0 * q0 + q1 * q1 + q2 * q2) - gb[c];
    }
    __syncthreads();

    const int lane  = tid & 31;
    const int half  = lane >> 4;
    const int l16   = lane & 15;
    const int itile = blockIdx.x * 8 + (tid >> 5);
    const int rowBase = itile * 16;

    const float* xb = X + (size_t)b * N * 3;
    const float* fb = F + (size_t)b * N;

    v2f a;
    {
        int r = rowBase + l16;
        float p0 = xb[r * 3 + 0], p1 = xb[r * 3 + 1], p2 = xb[r * 3 + 2];
        a.x = half ? p2 : p0;
        a.y = half ? 1.0f : p1;
    }
    float rr[8], fr[8];
#pragma unroll
    for (int k = 0; k < 8; ++k) {
        int r = rowBase + k + 8 * half;
        float p0 = xb[r * 3 + 0], p1 = xb[r * 3 + 1], p2 = xb[r * 3 + 2];
        rr[k] = p0 * p0 + p1 * p1 + p2 * p2;
        fr[k] = fb[r];
    }

    float acc = 0.0f;
    for (int jt = 0; jt < 128; ++jt) {
        int c = jt * 16 + l16;
        v2f bb;
        bb.x = lcol[c * 4 + 2 * half + 0];
        bb.y = lcol[c * 4 + 2 * half + 1];
        float gj = gb[c];
        v8f d = {};
        d = __builtin_amdgcn_wmma_f32_16x16x4_f32(false, a, false, bb,
                                                  (short)0, d, false, false);
#pragma unroll
        for (int k = 0; k < 8; ++k) {
            float S = d[k];                      // = C - g_j - |x_i|^2
            float C = S + gj + rr[k];
            float e = (fr[k] - rr[k] - S) * INV_EPS;   // (f_i + g_j - C)/EPS
            acc += __expf(e) * C;
        }
    }

    atomicAdd(&blockAcc, acc);      // LDS atomic (ds_add_f32)
    __syncthreads();
    if (tid == 0) atomicAdd(&out[b], blockAcc);
}

extern "C" void kernel_launch(void* const* d_in, const int* in_sizes, int n_in,
                              void* d_out, int out_size, void* d_ws, size_t ws_size,
                              hipStream_t stream) {
    const float* x = (const float*)d_in[0];   // [16,2048,3] f32
    const float* y = (const float*)d_in[1];   // [16,2048,3] f32
    float* out = (float*)d_out;               // [16] f32
    float* f = (float*)d_ws;                  // [16,2048]
    float* g = f + BATCH * N;                 // [16,2048]  (ws use: 256 KB)

    emd_prep<<<(BATCH * N + 255) / 256, 256, 0, stream>>>(f, g, out);

    dim3 grid(16, BATCH);   // 16 row-tile groups (8 tiles each) x 16 batches
    for (int it = 0; it < N_ITERS; ++it) {
        sinkhorn_half<<<grid, 256, 0, stream>>>(x, y, g, f);  // f-update
        sinkhorn_half<<<grid, 256, 0, stream>>>(y, x, f, g);  // g-update
    }
    emd_final<<<grid, 256, 0, stream>>>(x, y, f, g, out);
}